// BOCPD_GPTS_88613765251612
// MI455X (gfx1250) — compile-verified
//
#include <hip/hip_runtime.h>
#include <hip/hip_bf16.h>
#include <math.h>

#define T_N      2048
#define K_N      32
#define W_N      31
#define D_N      8
#define TP1      2049
#define HAZ_F    0.01f
#define TWO_PI_F 6.28318530717958647692f

typedef __attribute__((ext_vector_type(2))) float v2f;
typedef __attribute__((ext_vector_type(8))) float v8f;

// ---------------------------------------------------------------------------
// Kernel 1: one wave32 per timestep t0.
//  - reversed-order window (row i = X[t0-1-i]), row 31 = query X[t0]
//  - Gram G = Xr * Xr^T via V_WMMA_F32_16X16X4_F32 (K=8 -> 2 chained WMMAs/tile)
//  - build 31x31 RBF matrix (invalid tail = identity), one Cholesky,
//    two forward solves, prefix sums -> p[t0][0..31]
// ---------------------------------------------------------------------------
__global__ __launch_bounds__(32)
void gpts_upm_kernel(const float* __restrict__ X, const float* __restrict__ Y,
                     const float* __restrict__ p_ln, const float* __restrict__ p_ll,
                     const float* __restrict__ p_lv, float* __restrict__ pws)
{
    __shared__ float Xr[32][8];
    __shared__ float G[32][33];
    __shared__ float Am[32][33];
    __shared__ float vs[32];
    __shared__ float us[32];

    const int t0   = blockIdx.x;
    const int lane = threadIdx.x;
    const int nv   = (t0 < W_N) ? t0 : W_N;   // valid context length

    const float noise = expf(p_ln[0]);
    const float ls2   = expf(2.0f * p_ll[0]);
    const float var   = expf(p_lv[0]);

    // ---- load reversed window; invalid rows zeroed; row 31 = query point
    {
        const bool ok  = (lane == 31) || (lane < nv);
        const int  src = ok ? ((lane == 31) ? t0 : (t0 - 1 - lane)) : 0;
        #pragma unroll
        for (int d = 0; d < D_N; ++d)
            Xr[lane][d] = ok ? X[src * D_N + d] : 0.0f;
    }
    __syncthreads();

    // ---- G = Xr (32x8) * Xr^T via 4 tiles of f32 WMMA 16x16x4
    #pragma unroll
    for (int I = 0; I < 2; ++I) {
        #pragma unroll
        for (int J = 0; J < 2; ++J) {
            const int khalf = (lane < 16) ? 0 : 2;     // lanes 0-15: K=0,1 ; 16-31: K=2,3
            const int ra = I * 16 + (lane & 15);
            const int rb = J * 16 + (lane & 15);
            v2f a0, a1, b0, b1;
            a0.x = Xr[ra][0 + khalf]; a0.y = Xr[ra][1 + khalf];
            a1.x = Xr[ra][4 + khalf]; a1.y = Xr[ra][5 + khalf];
            b0.x = Xr[rb][0 + khalf]; b0.y = Xr[rb][1 + khalf];
            b1.x = Xr[rb][4 + khalf]; b1.y = Xr[rb][5 + khalf];
            v8f c = {};
            c = __builtin_amdgcn_wmma_f32_16x16x4_f32(false, a0, false, b0, (short)0, c, false, false);
            c = __builtin_amdgcn_wmma_f32_16x16x4_f32(false, a1, false, b1, (short)0, c, false, false);
            const int row0 = I * 16 + ((lane >= 16) ? 8 : 0);
            const int col  = J * 16 + (lane & 15);
            #pragma unroll
            for (int r = 0; r < 8; ++r)
                G[row0 + r][col] = c[r];
        }
    }
    __syncthreads();

    // ---- build RBF matrix in reversed order; tail (>= nv) = identity
    if (lane < W_N) {
        const int j = lane;
        for (int i = 0; i < W_N; ++i) {
            float a;
            if (i < nv && j < nv) {
                const float d2 = G[i][i] + G[j][j] - 2.0f * G[i][j];
                a = var * expf(-0.5f * d2 / ls2);
                if (i == j) a += noise;
            } else {
                a = (i == j) ? 1.0f : 0.0f;
            }
            Am[i][j] = a;
        }
    }

    // ---- rhs: kstar (from Gram row vs query) and windowed Y, zero past nv
    float b1 = 0.0f, b2 = 0.0f;
    if (lane < nv) {
        const float ds2 = G[lane][lane] + G[31][31] - 2.0f * G[lane][31];
        b1 = var * expf(-0.5f * ds2 / ls2);
        b2 = Y[t0 - 1 - lane];
    }
    __syncthreads();

    // ---- in-place Cholesky (lane = row), right-looking rank-1 updates
    for (int k = 0; k < W_N; ++k) {
        __syncthreads();
        const float dk = sqrtf(Am[k][k]);
        float lik = 0.0f;
        if (lane > k && lane < W_N) lik = Am[lane][k] / dk;
        __syncthreads();
        if (lane == k) Am[k][k] = dk;
        if (lane > k && lane < W_N) Am[lane][k] = lik;
        __syncthreads();
        if (lane > k && lane < W_N) {
            for (int j = k + 1; j <= lane; ++j)
                Am[lane][j] -= lik * Am[j][k];
        }
    }
    __syncthreads();

    // ---- forward solves L v = kstar, L u = yw (column-oriented, 2 RHS)
    for (int i = 0; i < W_N; ++i) {
        const float Lii = Am[i][i];
        if (lane == i) { vs[i] = b1 / Lii; us[i] = b2 / Lii; }
        __syncthreads();
        const float vi = vs[i], ui = us[i];
        if (lane > i && lane < W_N) {
            const float lji = Am[lane][i];
            b1 -= lji * vi;
            b2 -= lji * ui;
        }
        __syncthreads();
    }

    // ---- prefix sums give every context length L at once; write p row
    const float yt = Y[t0];
    if (lane < W_N) {
        float sm = 0.0f, sv = 0.0f;
        for (int i = 0; i <= lane; ++i) {
            sm += vs[i] * us[i];
            sv += vs[i] * vs[i];
        }
        const float fvar  = var + noise - sv;
        const float resid = yt - sm;
        pws[t0 * K_N + 1 + lane] =
            expf(-0.5f * (logf(TWO_PI_F * fvar) + resid * resid / fvar));
    } else {
        pws[t0 * K_N + 0] =
            expf(-0.5f * (logf(TWO_PI_F * var) + yt * yt / var));
    }
}

// ---------------------------------------------------------------------------
// Kernel 2: sequential BOCPD recursion in one 256-thread workgroup.
// Run-length vector lives in LDS.  sum(r_new) == sum(w), so a single
// reduction per step suffices; r_new[0] == HAZ exactly after normalization.
// ---------------------------------------------------------------------------
__global__ __launch_bounds__(256)
void bocpd_kernel(const float* __restrict__ p, float* __restrict__ out)
{
    __shared__ float rvec[TP1];
    __shared__ float wbuf[TP1];
    __shared__ float pl[K_N];
    __shared__ float sred[8];
    const int tid = threadIdx.x;

    // t = 0 column: r0 = e_0
    for (int r = tid; r < TP1; r += 256) {
        const float v = (r == 0) ? 1.0f : 0.0f;
        rvec[r] = v;
        out[(size_t)r * TP1 + 0] = v;
    }
    __syncthreads();

    for (int t = 1; t <= T_N; ++t) {
        if (tid < K_N) pl[tid] = p[(t - 1) * K_N + tid];
        __syncthreads();

        float part = 0.0f;
        for (int r = tid; r < TP1; r += 256) {
            const float w = (r < t) ? rvec[r] * pl[(r < 31) ? r : 31] : 0.0f;
            wbuf[r] = w;
            part += w;
        }
        #pragma unroll
        for (int off = 16; off > 0; off >>= 1)
            part += __shfl_down(part, off, 32);
        if ((tid & 31) == 0) sred[tid >> 5] = part;
        __syncthreads();

        float S = 0.0f;
        #pragma unroll
        for (int i = 0; i < 8; ++i) S += sred[i];
        const float inv = (1.0f - HAZ_F) / S;

        for (int r = tid; r < TP1; r += 256) {
            const float v = (r == 0) ? HAZ_F : wbuf[r - 1] * inv;
            rvec[r] = v;
            out[(size_t)r * TP1 + t] = v;
        }
        __syncthreads();
    }
}

// ---------------------------------------------------------------------------
extern "C" void kernel_launch(void* const* d_in, const int* in_sizes, int n_in,
                              void* d_out, int out_size, void* d_ws, size_t ws_size,
                              hipStream_t stream)
{
    const float* X  = (const float*)d_in[0];   // (T, D) f32
    const float* Y  = (const float*)d_in[1];   // (T,)   f32
    const float* ln = (const float*)d_in[2];   // log_noise
    const float* ll = (const float*)d_in[3];   // log_lengthscale
    const float* lv = (const float*)d_in[4];   // log_variance

    float* pws = (float*)d_ws;                 // T_N * 32 floats (256 KB)
    float* out = (float*)d_out;                // (T+1)*(T+1) f32

    gpts_upm_kernel<<<T_N, 32, 0, stream>>>(X, Y, ln, ll, lv, pws);
    bocpd_kernel<<<1, 256, 0, stream>>>(pws, out);
}